// SimpleOrbitalAEVComputer_74921409511786
// MI455X (gfx1250) — compile-verified
//
#include <hip/hip_runtime.h>
#include <math.h>

typedef __attribute__((ext_vector_type(2))) float v2f;
typedef __attribute__((ext_vector_type(4))) float v4f;
typedef __attribute__((ext_vector_type(8))) float v8f;

// Packed (M<<4)|N pair tables for the 12x12 angle matrix.
// UPPER66: strict upper triangle (M<N) -> needs acos.  LOWER78: M>=N -> zero.
__constant__ unsigned char UPPER66[66] = {
    0x01,0x02,0x03,0x04,0x05,0x06,0x07,0x08,0x09,0x0A,0x0B,
    0x12,0x13,0x14,0x15,0x16,0x17,0x18,0x19,0x1A,0x1B,
    0x23,0x24,0x25,0x26,0x27,0x28,0x29,0x2A,0x2B,
    0x34,0x35,0x36,0x37,0x38,0x39,0x3A,0x3B,
    0x45,0x46,0x47,0x48,0x49,0x4A,0x4B,
    0x56,0x57,0x58,0x59,0x5A,0x5B,
    0x67,0x68,0x69,0x6A,0x6B,
    0x78,0x79,0x7A,0x7B,
    0x89,0x8A,0x8B,
    0x9A,0x9B,
    0xAB};

__constant__ unsigned char LOWER78[78] = {
    0x00,
    0x10,0x11,
    0x20,0x21,0x22,
    0x30,0x31,0x32,0x33,
    0x40,0x41,0x42,0x43,0x44,
    0x50,0x51,0x52,0x53,0x54,0x55,
    0x60,0x61,0x62,0x63,0x64,0x65,0x66,
    0x70,0x71,0x72,0x73,0x74,0x75,0x76,0x77,
    0x80,0x81,0x82,0x83,0x84,0x85,0x86,0x87,0x88,
    0x90,0x91,0x92,0x93,0x94,0x95,0x96,0x97,0x98,0x99,
    0xA0,0xA1,0xA2,0xA3,0xA4,0xA5,0xA6,0xA7,0xA8,0xA9,0xAA,
    0xB0,0xB1,0xB2,0xB3,0xB4,0xB5,0xB6,0xB7,0xB8,0xB9,0xBA,0xBB};

// One wave32 per atom. 12x3 orbital matrix (padded to 16x4) -> Gram via
// V_WMMA_F32_16X16X4_F32 (A/B share registers: B's 4x16 column-major striping
// mirrors A's 16x4 row striping). C is staged to LDS so the 66 upper-triangle
// acos evaluations compact into ceil(66/32)=3 rounds instead of 8.
__global__ __launch_bounds__(256) void SimpleOrbitalAEV_kernel(
    const float* __restrict__ in,   // (n_atoms, 45)
    float* __restrict__ out,        // (n_atoms, 165)
    int n_atoms)
{
    __shared__ float smem[8][256];          // 16x16 f32 tile per wave (8 KB)

    const int lane   = threadIdx.x & 31;
    const int waveId = threadIdx.x >> 5;
    const long long atom = (long long)blockIdx.x * 8 + waveId;
    if (atom >= n_atoms) return;            // wave-uniform guard

    const long long in_base  = atom * 45;
    const long long out_base = atom * 165;

    const int half = lane >> 4;   // 0 -> K=0,1 ; 1 -> K=2,(3=zero pad)
    const int col  = lane & 15;   // orbital row for A, column N for C
    const int Mr   = (col < 12) ? col : 11;   // clamp for in-bounds addressing

    // Component offsets within the 45-float record for orbital row Mr.
    // Rows 0..3: p at 9+3*M.  Rows 4..11: d, reordered by D_REORDER =
    // {0,2,5,4,3,1}, packed as nibbles into 0x134520.
    const int r    = Mr - 4;
    const int jb   = 3 * (r & 1);
    const int dbas = 21 + 6 * (r >> 1);
    const unsigned DR = 0x134520u;
    const int pbas = 9 + 3 * Mr;
    const bool isP = (Mr < 4);
    const int off0 = isP ? (pbas + 0) : (dbas + (int)((DR >> (4 * (jb + 0))) & 0xF));
    const int off1 = isP ? (pbas + 1) : (dbas + (int)((DR >> (4 * (jb + 1))) & 0xF));
    const int off2 = isP ? (pbas + 2) : (dbas + (int)((DR >> (4 * (jb + 2))) & 0xF));

    float x = in[in_base + off0];
    float y = in[in_base + off1];
    float z = in[in_base + off2];
    const bool valid = (col < 12);
    x = valid ? x : 0.0f;
    y = valid ? y : 0.0f;
    z = valid ? z : 0.0f;

    const float s2  = x * x + y * y + z * z;
    const float inv = valid ? __builtin_amdgcn_rsqf(s2) : 0.0f;  // v_rsq_f32
    const float dist = s2 * inv;          // sqrt(s2) via the rsq already in hand
    const float nx = x * inv, ny = y * inv, nz = z * inv;

    // A-operand (16x4 f32, 2 VGPRs): lanes 0-15 hold {K0,K1}, lanes 16-31 {K2,0}.
    v2f a;
    a.x = half ? nz : nx;
    a.y = half ? 0.0f : ny;

    // Gram = A * A^T  (same registers feed SRC0 and SRC1).
    v8f c = {};
    c = __builtin_amdgcn_wmma_f32_16x16x4_f32(
        /*neg_a=*/false, a, /*neg_b=*/false, a,
        /*c_mod=*/(short)0, c, /*reuse_a=*/false, /*reuse_b=*/false);

    // Stage C into LDS column-major: L[N*16 + M]. Lane owns column N=col,
    // rows 8*half+0..7 -> contiguous, so two ds_store_b128 cover it.
    float* L = smem[waveId];
    const int cb = col * 16 + 8 * half;
    v4f lo = {c[0], c[1], c[2], c[3]};
    v4f hi = {c[4], c[5], c[6], c[7]};
    *(v4f*)(L + cb)     = lo;
    *(v4f*)(L + cb + 4) = hi;
    asm volatile("" ::: "memory");   // keep gathers after stores (same-wave LDS is in-order)

    // --- outputs: [s(9) | distances(12) | angles(12x12 row-major, 144)] ---
    if (lane < 9)
        out[out_base + lane] = in[in_base + lane];

    if (half == 0 && col < 12)
        out[out_base + 9 + col] = dist;

    // 66 upper-triangle angles in 3 compact rounds (one acos per round).
#pragma unroll
    for (int t = 0; t < 3; ++t) {
        const int k  = t * 32 + lane;
        const int pk = UPPER66[(k < 66) ? k : 65];
        const int Mg = pk >> 4;
        const int Ng = pk & 15;
        const float g   = L[Ng * 16 + Mg];
        const float val = acosf(0.95f * g);
        if (k < 66)
            out[out_base + 21 + Mg * 12 + Ng] = val;
    }

    // 78 zero slots (diagonal + lower triangle), disjoint from the acos stores.
#pragma unroll
    for (int t = 0; t < 3; ++t) {
        const int k = t * 32 + lane;
        if (k < 78) {
            const int pk = LOWER78[k];
            out[out_base + 21 + (pk >> 4) * 12 + (pk & 15)] = 0.0f;
        }
    }
}

extern "C" void kernel_launch(void* const* d_in, const int* in_sizes, int n_in,
                              void* d_out, int out_size, void* d_ws, size_t ws_size,
                              hipStream_t stream) {
    const float* coeff = (const float*)d_in[0];
    float* out = (float*)d_out;

    const int n_atoms = in_sizes[0] / 45;          // 4096 * 128 = 524288
    const int waves_per_block = 8;                 // 256 threads, wave32
    const int blocks = (n_atoms + waves_per_block - 1) / waves_per_block;

    SimpleOrbitalAEV_kernel<<<blocks, 256, 0, stream>>>(coeff, out, n_atoms);
}